// MessagePassing_38328288149875
// MI455X (gfx1250) — compile-verified
//
#include <hip/hip_runtime.h>
#include <hip/hip_bf16.h>

typedef float v2f __attribute__((ext_vector_type(2)));
typedef float v8f __attribute__((ext_vector_type(8)));

constexpr int NB = 8;       // batch
constexpr int NA = 128;     // atoms
constexpr int NF = 64;      // features
constexpr int NK = 16;      // radial basis
constexpr int NL = 9;       // irreps
constexpr int NE = 16256;   // edges (128*127)
constexpr int NET = NE / 16;       // 1016 edge tiles
constexpr float SQ3F = 1.7320508075688772f;

__constant__ float c_BIN[16] = {1.f,15.f,105.f,455.f,1365.f,3003.f,5005.f,6435.f,
                                6435.f,5005.f,3003.f,1365.f,455.f,105.f,15.f,1.f};

__device__ __forceinline__ int degf(int l) { return l == 0 ? 0 : (l <= 3 ? 1 : 2); }
__device__ __forceinline__ float sigmf_(float x) { return 1.f / (1.f + expf(-x)); }

// ---------------------------------------------------------------------------
// Wave-level GEMM: C[16 x 64] += A[16 x K] * B[K x 64] using V_WMMA_F32_16X16X4_F32
// A element (m,k) at A[m*lda + k]   (k-stride always 1 in this app)
// optional addend A2 with its own row stride
// B element (k,n) at Bm[k*ldbk + n*ldbn]   (ldbn=NF,ldbk=1 gives B^T reads)
// acc[t] covers output columns t*16..t*16+15.
// C/D layout: acc[t][v] -> row = v + 8*(lane/16), col = t*16 + lane%16
// ---------------------------------------------------------------------------
template <bool HASA2>
__device__ __forceinline__ void gemm16(const float* __restrict__ A, int lda,
                                       const float* __restrict__ A2, int lda2,
                                       const float* __restrict__ Bm, int ldbk, int ldbn,
                                       int K, v8f acc[4]) {
  const int lane = threadIdx.x & 31;
  const int lo = lane & 15, hi = lane >> 4;
  for (int k0 = 0; k0 < K; k0 += 4) {
    const int ka = k0 + 2 * hi;
    v2f av;
    av.x = A[lo * lda + ka];
    av.y = A[lo * lda + ka + 1];
    if (HASA2) {
      av.x += A2[lo * lda2 + ka];
      av.y += A2[lo * lda2 + ka + 1];
    }
#pragma unroll
    for (int t = 0; t < 4; ++t) {
      const int n = t * 16 + lo;
      v2f bv;
      bv.x = Bm[ka * ldbk + n * ldbn];
      bv.y = Bm[(ka + 1) * ldbk + n * ldbn];
      acc[t] = __builtin_amdgcn_wmma_f32_16x16x4_f32(false, av, false, bv,
                                                     (short)0, acc[t], false, false);
    }
  }
}

// ---------------------------------------------------------------------------
// Edge basis: Y[e,9], R[e,16] = rad*cut
// ---------------------------------------------------------------------------
__global__ void __launch_bounds__(256) k_edge_basis(
    const float* __restrict__ pos, const int* __restrict__ srcI,
    const int* __restrict__ dstI, float* __restrict__ Ybuf, float* __restrict__ Rbuf) {
  int idx = blockIdx.x * blockDim.x + threadIdx.x;
  if (idx >= NB * NE) return;
  int b = idx / NE, e = idx % NE;
  int s = srcI[e], d = dstI[e];
  const float* pb = pos + (size_t)b * NA * 3;
  float dx = pb[s * 3 + 0] - pb[d * 3 + 0];
  float dy = pb[s * 3 + 1] - pb[d * 3 + 1];
  float dz = pb[s * 3 + 2] - pb[d * 3 + 2];
  float r = sqrtf(dx * dx + dy * dy + dz * dz + 1e-12f);
  float inv = 1.f / r;
  float ux = dx * inv, uy = dy * inv, uz = dz * inv;
  float* Yo = Ybuf + (size_t)idx * NL;
  Yo[0] = 1.f; Yo[1] = ux; Yo[2] = uy; Yo[3] = uz;
  Yo[4] = SQ3F * ux * uy; Yo[5] = SQ3F * uy * uz;
  Yo[6] = 0.5f * (3.f * uz * uz - 1.f);
  Yo[7] = SQ3F * ux * uz; Yo[8] = 0.5f * SQ3F * (ux * ux - uy * uy);
  float w = 1.f / (1.f + r), om = 1.f - w;
  float t = r * 0.2f; t = t * t;
  float cut = 0.f;
  if (t < 1.f) { float den = fmaxf(1.f - t, 1e-12f); cut = expf(1.f - 1.f / den); }
  float wp[16], op[16];
  wp[0] = 1.f; op[0] = 1.f;
#pragma unroll
  for (int k = 1; k < 16; ++k) { wp[k] = wp[k - 1] * w; op[k] = op[k - 1] * om; }
  float* Ro = Rbuf + (size_t)idx * NK;
#pragma unroll
  for (int k = 0; k < 16; ++k) Ro[k] = c_BIN[k] * wp[k] * op[15 - k] * cut;
}

// X0 init: x[:,0,:] = embed[z], rest 0
__global__ void __launch_bounds__(256) k_embed(
    const int* __restrict__ z, const float* __restrict__ embed, float* __restrict__ X0) {
  int idx = blockIdx.x * blockDim.x + threadIdx.x;
  if (idx >= NB * NA * NL * NF) return;
  int f = idx % NF;
  int l = (idx / NF) % NL;
  int n = idx / (NF * NL);   // b*NA + atom
  X0[idx] = (l == 0) ? embed[z[n] * NF + f] : 0.f;
}

// ---------------------------------------------------------------------------
// Interaction layer forward message: PA/PB via WMMA, scatter into ysum
// one wave per (b, edge-tile)
// ---------------------------------------------------------------------------
__global__ void __launch_bounds__(128, 1) k_msg_fwd(
    const float* __restrict__ Rbuf, const float* __restrict__ Ybuf,
    const float* __restrict__ X, const float* __restrict__ WAi,
    const float* __restrict__ WBi, const int* __restrict__ srcI,
    const int* __restrict__ dstI, float* __restrict__ ysum) {
  int wid = (blockIdx.x * blockDim.x + threadIdx.x) >> 5;
  int lane = threadIdx.x & 31, lo = lane & 15, hi = lane >> 4;
  int b = wid / NET, et = wid % NET;
  const float* Rb = Rbuf + ((size_t)b * NE + et * 16) * NK;
  v8f accA[4] = {}; v8f accB[4] = {};
  gemm16<false>(Rb, NK, nullptr, 0, WAi, NF, 1, NK, accA);
  gemm16<false>(Rb, NK, nullptr, 0, WBi, NF, 1, NK, accB);
  const float* Xb = X + (size_t)b * NA * NL * NF;
  float* ysb = ysum + (size_t)b * NA * NL * NF;
  const float* Yb = Ybuf + ((size_t)b * NE + et * 16) * NL;
#pragma unroll
  for (int v = 0; v < 8; ++v) {
    int er = v + 8 * hi;
    int e = et * 16 + er;
    int s = srcI[e], d = dstI[e];
    float xs0[4];
#pragma unroll
    for (int t = 0; t < 4; ++t) xs0[t] = Xb[(s * NL) * NF + t * 16 + lo];
#pragma unroll
    for (int l = 0; l < 9; ++l) {
      float yl = Yb[er * NL + l];
#pragma unroll
      for (int t = 0; t < 4; ++t) {
        int f = t * 16 + lo;
        float msg = yl * accA[t][v] * xs0[t] + Xb[(s * NL + l) * NF + f] * accB[t][v];
        atomicAdd(&ysb[(d * NL + l) * NF + f], msg);
      }
    }
  }
}

// y1 = (x + ysum) @ W1[deg(l)]  (+ b1 on l==0); one wave per (b,l,mtile)
__global__ void __launch_bounds__(128, 1) k_atom_fwd1(
    const float* __restrict__ X, const float* __restrict__ ysum,
    const float* __restrict__ W1i, const float* __restrict__ b1i, float* __restrict__ Y1) {
  int wid = (blockIdx.x * blockDim.x + threadIdx.x) >> 5;
  int lane = threadIdx.x & 31, lo = lane & 15, hi = lane >> 4;
  int mt = wid & 7, l = (wid >> 3) % NL, b = wid / (8 * NL);
  size_t base = ((size_t)b * NA + mt * 16) * NL * NF + l * NF;
  const float* A = X + base;
  const float* A2 = ysum + base;
  const float* Bm = W1i + degf(l) * NF * NF;
  v8f acc[4] = {};
  gemm16<true>(A, NL * NF, A2, NL * NF, Bm, NF, 1, NF, acc);
#pragma unroll
  for (int t = 0; t < 4; ++t)
#pragma unroll
    for (int v = 0; v < 8; ++v) {
      int g = t * 16 + lo, row = mt * 16 + v + 8 * hi;
      float val = acc[t][v] + (l == 0 ? b1i[g] : 0.f);
      Y1[((size_t)b * NA + row) * NL * NF + l * NF + g] = val;
    }
}

// gating: out0 = silu(s); out_l = y1_l * sigmoid(s)
__global__ void __launch_bounds__(256) k_gate(const float* __restrict__ Y1,
                                              float* __restrict__ G) {
  int idx = blockIdx.x * blockDim.x + threadIdx.x;
  if (idx >= NB * NA * NF) return;
  int f = idx % NF, n = idx / NF;
  size_t base = (size_t)n * NL * NF + f;
  float s = Y1[base];
  float sg = sigmf_(s);
  G[base] = s * sg;
#pragma unroll
  for (int l = 1; l < 9; ++l) G[base + l * NF] = Y1[base + l * NF] * sg;
}

// x_next = x + G @ W2[deg(l)]
__global__ void __launch_bounds__(128, 1) k_atom_fwd2(
    const float* __restrict__ G, const float* __restrict__ X,
    const float* __restrict__ W2i, float* __restrict__ Xn) {
  int wid = (blockIdx.x * blockDim.x + threadIdx.x) >> 5;
  int lane = threadIdx.x & 31, lo = lane & 15, hi = lane >> 4;
  int mt = wid & 7, l = (wid >> 3) % NL, b = wid / (8 * NL);
  size_t base = ((size_t)b * NA + mt * 16) * NL * NF + l * NF;
  const float* Bm = W2i + degf(l) * NF * NF;
  v8f acc[4] = {};
  gemm16<false>(G + base, NL * NF, nullptr, 0, Bm, NF, 1, NF, acc);
#pragma unroll
  for (int t = 0; t < 4; ++t)
#pragma unroll
    for (int v = 0; v < 8; ++v) {
      int g = t * 16 + lo, row = mt * 16 + v + 8 * hi;
      size_t o = ((size_t)b * NA + row) * NL * NF + l * NF + g;
      Xn[o] = X[o] + acc[t][v];
    }
}

// final readout message: mA = PA2*xs0, mB = PB2 * sum_l Y_l xs_l -> ysum0
__global__ void __launch_bounds__(128, 1) k_msg_final_fwd(
    const float* __restrict__ Rbuf, const float* __restrict__ Ybuf,
    const float* __restrict__ X2, const float* __restrict__ WA2,
    const float* __restrict__ WB2, const int* __restrict__ srcI,
    const int* __restrict__ dstI, float* __restrict__ ysum0) {
  int wid = (blockIdx.x * blockDim.x + threadIdx.x) >> 5;
  int lane = threadIdx.x & 31, lo = lane & 15, hi = lane >> 4;
  int b = wid / NET, et = wid % NET;
  const float* Rb = Rbuf + ((size_t)b * NE + et * 16) * NK;
  v8f accA[4] = {}; v8f accB[4] = {};
  gemm16<false>(Rb, NK, nullptr, 0, WA2, NF, 1, NK, accA);
  gemm16<false>(Rb, NK, nullptr, 0, WB2, NF, 1, NK, accB);
  const float* Xb = X2 + (size_t)b * NA * NL * NF;
  float* ysb = ysum0 + (size_t)b * NA * NF;
  const float* Yb = Ybuf + ((size_t)b * NE + et * 16) * NL;
#pragma unroll
  for (int v = 0; v < 8; ++v) {
    int er = v + 8 * hi;
    int e = et * 16 + er;
    int s = srcI[e], d = dstI[e];
    float Yl[9];
#pragma unroll
    for (int l = 0; l < 9; ++l) Yl[l] = Yb[er * NL + l];
#pragma unroll
    for (int t = 0; t < 4; ++t) {
      int f = t * 16 + lo;
      float xs0 = Xb[(s * NL) * NF + f];
      float S = 0.f;
#pragma unroll
      for (int l = 0; l < 9; ++l) S += Yl[l] * Xb[(s * NL + l) * NF + f];
      atomicAdd(&ysb[d * NF + f], accA[t][v] * xs0 + accB[t][v] * S);
    }
  }
}

// a = (x2[:,0,:] + ysum0) @ W1s + b1s; one wave per (b, mtile)
__global__ void __launch_bounds__(128, 1) k_mlp1(
    const float* __restrict__ X2, const float* __restrict__ ysum0,
    const float* __restrict__ W1s, const float* __restrict__ b1s,
    float* __restrict__ Abuf) {
  int wid = (blockIdx.x * blockDim.x + threadIdx.x) >> 5;
  int lane = threadIdx.x & 31, lo = lane & 15, hi = lane >> 4;
  int mt = wid & 7, b = wid >> 3;
  const float* A = X2 + ((size_t)b * NA + mt * 16) * NL * NF;
  const float* A2 = ysum0 + ((size_t)b * NA + mt * 16) * NF;
  v8f acc[4] = {};
  gemm16<true>(A, NL * NF, A2, NF, W1s, NF, 1, NF, acc);
#pragma unroll
  for (int t = 0; t < 4; ++t)
#pragma unroll
    for (int v = 0; v < 8; ++v) {
      int g = t * 16 + lo, row = mt * 16 + v + 8 * hi;
      Abuf[((size_t)b * NA + row) * NF + g] = acc[t][v] + b1s[g];
    }
}

__global__ void __launch_bounds__(256) k_silu(const float* __restrict__ Abuf,
                                              float* __restrict__ H1) {
  int idx = blockIdx.x * blockDim.x + threadIdx.x;
  if (idx >= NB * NA * NF) return;
  float a = Abuf[idx];
  H1[idx] = a * sigmf_(a);
}

// h2 = H1 @ W2s ; x0out = x2[:,0,:] + h2 ; E += x0out . w_out + bias[z]
__global__ void __launch_bounds__(128, 1) k_mlp2(
    const float* __restrict__ H1, const float* __restrict__ W2s,
    const float* __restrict__ X2, const float* __restrict__ w_out,
    const float* __restrict__ ebias, const int* __restrict__ z,
    float* __restrict__ Eout) {
  int wid = (blockIdx.x * blockDim.x + threadIdx.x) >> 5;
  int lane = threadIdx.x & 31, lo = lane & 15, hi = lane >> 4;
  int mt = wid & 7, b = wid >> 3;
  const float* A = H1 + ((size_t)b * NA + mt * 16) * NF;
  v8f acc[4] = {};
  gemm16<false>(A, NF, nullptr, 0, W2s, NF, 1, NF, acc);
#pragma unroll
  for (int v = 0; v < 8; ++v) {
    int row = mt * 16 + v + 8 * hi;
    float part = 0.f;
#pragma unroll
    for (int t = 0; t < 4; ++t) {
      int g = t * 16 + lo;
      float x0o = X2[((size_t)b * NA + row) * NL * NF + g] + acc[t][v];
      part += x0o * w_out[g];
    }
    part += __shfl_xor(part, 1, 32);
    part += __shfl_xor(part, 2, 32);
    part += __shfl_xor(part, 4, 32);
    part += __shfl_xor(part, 8, 32);
    if (lo == 0) atomicAdd(&Eout[b], part + ebias[z[b * NA + row]]);
  }
}

// gh1[f] = sum_g w_out[g] * W2s[f,g]
__global__ void k_gh1(const float* __restrict__ w_out, const float* __restrict__ W2s,
                      float* __restrict__ gh1) {
  int f = threadIdx.x;
  if (f >= NF) return;
  float a = 0.f;
  for (int g = 0; g < NF; ++g) a += w_out[g] * W2s[f * NF + g];
  gh1[f] = a;
}

// g_a = gh1[f] * silu'(a)
__global__ void __launch_bounds__(256) k_gax(const float* __restrict__ Abuf,
                                             const float* __restrict__ gh1,
                                             float* __restrict__ GA) {
  int idx = blockIdx.x * blockDim.x + threadIdx.x;
  if (idx >= NB * NA * NF) return;
  float a = Abuf[idx];
  float sg = sigmf_(a);
  GA[idx] = gh1[idx % NF] * sg * (1.f + a * (1.f - sg));
}

// g_h0 = GA @ W1s^T ; gX2[:,0,:] = w_out + g_h0 ; Gh0 = g_h0
__global__ void __launch_bounds__(128, 1) k_mlp_bwd(
    const float* __restrict__ GA, const float* __restrict__ W1s,
    const float* __restrict__ w_out, float* __restrict__ Gh0,
    float* __restrict__ gX2) {
  int wid = (blockIdx.x * blockDim.x + threadIdx.x) >> 5;
  int lane = threadIdx.x & 31, lo = lane & 15, hi = lane >> 4;
  int mt = wid & 7, b = wid >> 3;
  const float* A = GA + ((size_t)b * NA + mt * 16) * NF;
  v8f acc[4] = {};
  gemm16<false>(A, NF, nullptr, 0, W1s, 1, NF, NF, acc);   // B^T
#pragma unroll
  for (int t = 0; t < 4; ++t)
#pragma unroll
    for (int v = 0; v < 8; ++v) {
      int g = t * 16 + lo, row = mt * 16 + v + 8 * hi;
      float val = acc[t][v];
      Gh0[((size_t)b * NA + row) * NF + g] = val;
      gX2[((size_t)b * NA + row) * NL * NF + g] = w_out[g] + val;
    }
}

// final readout backward: edge-level adjoints, gX2 scatter, g_Y/g_R accumulate
__global__ void __launch_bounds__(128, 1) k_msg_final_bwd(
    const float* __restrict__ Rbuf, const float* __restrict__ Ybuf,
    const float* __restrict__ X2, const float* __restrict__ WA2,
    const float* __restrict__ WB2, const int* __restrict__ srcI,
    const int* __restrict__ dstI, const float* __restrict__ Gh0,
    float* __restrict__ gX2, float* __restrict__ gYe, float* __restrict__ gRe) {
  __shared__ float sPA[4][16][64];
  __shared__ float sPB[4][16][64];
  int wid = (blockIdx.x * blockDim.x + threadIdx.x) >> 5;
  int wslot = (threadIdx.x >> 5);
  int lane = threadIdx.x & 31, lo = lane & 15, hi = lane >> 4;
  int b = wid / NET, et = wid % NET;
  const float* Rb = Rbuf + ((size_t)b * NE + et * 16) * NK;
  v8f accA[4] = {}; v8f accB[4] = {};
  gemm16<false>(Rb, NK, nullptr, 0, WA2, NF, 1, NK, accA);
  gemm16<false>(Rb, NK, nullptr, 0, WB2, NF, 1, NK, accB);
  const float* Xb = X2 + (size_t)b * NA * NL * NF;
  float* gXb = gX2 + (size_t)b * NA * NL * NF;
  const float* Gh = Gh0 + (size_t)b * NA * NF;
  const float* Yb = Ybuf + ((size_t)b * NE + et * 16) * NL;
#pragma unroll
  for (int v = 0; v < 8; ++v) {
    int er = v + 8 * hi;
    int e = et * 16 + er;
    int s = srcI[e], d = dstI[e];
    float Yl[9];
#pragma unroll
    for (int l = 0; l < 9; ++l) Yl[l] = Yb[er * NL + l];
    float gm[4], xs0[4], gS[4], Sv[4];
#pragma unroll
    for (int t = 0; t < 4; ++t) {
      int f = t * 16 + lo;
      gm[t] = Gh[d * NF + f];
      xs0[t] = Xb[(s * NL) * NF + f];
      gS[t] = gm[t] * accB[t][v];
      Sv[t] = 0.f;
    }
    float gy[9];
#pragma unroll
    for (int l = 0; l < 9; ++l) {
      float gyl = 0.f;
#pragma unroll
      for (int t = 0; t < 4; ++t) {
        int f = t * 16 + lo;
        float xsl = Xb[(s * NL + l) * NF + f];
        Sv[t] += Yl[l] * xsl;
        gyl += gS[t] * xsl;
        float gxs = gS[t] * Yl[l] + (l == 0 ? gm[t] * accA[t][v] : 0.f);
        atomicAdd(&gXb[(s * NL + l) * NF + f], gxs);
      }
      gy[l] = gyl;
    }
#pragma unroll
    for (int t = 0; t < 4; ++t) {
      int f = t * 16 + lo;
      sPA[wslot][er][f] = gm[t] * xs0[t];
      sPB[wslot][er][f] = gm[t] * Sv[t];
    }
#pragma unroll
    for (int l = 0; l < 9; ++l) {
      float r = gy[l];
      r += __shfl_xor(r, 1, 32); r += __shfl_xor(r, 2, 32);
      r += __shfl_xor(r, 4, 32); r += __shfl_xor(r, 8, 32);
      if (lo == 0) gYe[((size_t)b * NE + e) * NL + l] += r;
    }
  }
  __syncthreads();
#pragma unroll
  for (int k8 = 0; k8 < 8; ++k8) {
    int k = hi * 8 + k8;
    float a = 0.f;
    for (int f = 0; f < NF; ++f)
      a += sPA[wslot][lo][f] * WA2[k * NF + f] + sPB[wslot][lo][f] * WB2[k * NF + f];
    gRe[((size_t)b * NE + et * 16 + lo) * NK + k] += a;
  }
}

// g_out = gXout @ W2^T
__global__ void __launch_bounds__(128, 1) k_atom_bwd2(
    const float* __restrict__ gXout, const float* __restrict__ W2i,
    float* __restrict__ G) {
  int wid = (blockIdx.x * blockDim.x + threadIdx.x) >> 5;
  int lane = threadIdx.x & 31, lo = lane & 15, hi = lane >> 4;
  int mt = wid & 7, l = (wid >> 3) % NL, b = wid / (8 * NL);
  size_t base = ((size_t)b * NA + mt * 16) * NL * NF + l * NF;
  const float* Bm = W2i + degf(l) * NF * NF;
  v8f acc[4] = {};
  gemm16<false>(gXout + base, NL * NF, nullptr, 0, Bm, 1, NF, NF, acc);   // B^T
#pragma unroll
  for (int t = 0; t < 4; ++t)
#pragma unroll
    for (int v = 0; v < 8; ++v) {
      int g = t * 16 + lo, row = mt * 16 + v + 8 * hi;
      G[((size_t)b * NA + row) * NL * NF + l * NF + g] = acc[t][v];
    }
}

// gating backward (in place on G)
__global__ void __launch_bounds__(256) k_gate_bwd(const float* __restrict__ Y1,
                                                  float* __restrict__ G) {
  int idx = blockIdx.x * blockDim.x + threadIdx.x;
  if (idx >= NB * NA * NF) return;
  int f = idx % NF, n = idx / NF;
  size_t base = (size_t)n * NL * NF + f;
  float s = Y1[base];
  float sg = sigmf_(s);
  float sp = sg * (1.f + s * (1.f - sg));
  float accv = G[base] * sp;
  float sgp = sg * (1.f - sg);
#pragma unroll
  for (int l = 1; l < 9; ++l) {
    float go = G[base + l * NF];
    accv += go * Y1[base + l * NF] * sgp;
    G[base + l * NF] = go * sg;
  }
  G[base] = accv;
}

// g_y = G @ W1^T ; GYb = g_y ; gXin = gXout + g_y
__global__ void __launch_bounds__(128, 1) k_atom_bwd1(
    const float* __restrict__ G, const float* __restrict__ W1i,
    const float* __restrict__ gXout, float* __restrict__ GYb,
    float* __restrict__ gXin) {
  int wid = (blockIdx.x * blockDim.x + threadIdx.x) >> 5;
  int lane = threadIdx.x & 31, lo = lane & 15, hi = lane >> 4;
  int mt = wid & 7, l = (wid >> 3) % NL, b = wid / (8 * NL);
  size_t base = ((size_t)b * NA + mt * 16) * NL * NF + l * NF;
  const float* Bm = W1i + degf(l) * NF * NF;
  v8f acc[4] = {};
  gemm16<false>(G + base, NL * NF, nullptr, 0, Bm, 1, NF, NF, acc);   // B^T
#pragma unroll
  for (int t = 0; t < 4; ++t)
#pragma unroll
    for (int v = 0; v < 8; ++v) {
      int g = t * 16 + lo, row = mt * 16 + v + 8 * hi;
      size_t o = ((size_t)b * NA + row) * NL * NF + l * NF + g;
      float val = acc[t][v];
      GYb[o] = val;
      gXin[o] = gXout[o] + val;
    }
}

// interaction layer message backward
__global__ void __launch_bounds__(128, 1) k_msg_bwd(
    const float* __restrict__ Rbuf, const float* __restrict__ Ybuf,
    const float* __restrict__ Xin, const float* __restrict__ WAi,
    const float* __restrict__ WBi, const int* __restrict__ srcI,
    const int* __restrict__ dstI, const float* __restrict__ GYb,
    float* __restrict__ gXin, float* __restrict__ gYe, float* __restrict__ gRe) {
  __shared__ float sPA[4][16][64];
  __shared__ float sPB[4][16][64];
  int wid = (blockIdx.x * blockDim.x + threadIdx.x) >> 5;
  int wslot = (threadIdx.x >> 5);
  int lane = threadIdx.x & 31, lo = lane & 15, hi = lane >> 4;
  int b = wid / NET, et = wid % NET;
  const float* Rb = Rbuf + ((size_t)b * NE + et * 16) * NK;
  v8f accA[4] = {}; v8f accB[4] = {};
  gemm16<false>(Rb, NK, nullptr, 0, WAi, NF, 1, NK, accA);
  gemm16<false>(Rb, NK, nullptr, 0, WBi, NF, 1, NK, accB);
  const float* Xb = Xin + (size_t)b * NA * NL * NF;
  float* gXb = gXin + (size_t)b * NA * NL * NF;
  const float* Gb = GYb + (size_t)b * NA * NL * NF;
  const float* Yb = Ybuf + ((size_t)b * NE + et * 16) * NL;
#pragma unroll
  for (int v = 0; v < 8; ++v) {
    int er = v + 8 * hi;
    int e = et * 16 + er;
    int s = srcI[e], d = dstI[e];
    float Yl[9];
#pragma unroll
    for (int l = 0; l < 9; ++l) Yl[l] = Yb[er * NL + l];
    float xs0[4], sYg[4], gPBv[4];
#pragma unroll
    for (int t = 0; t < 4; ++t) {
      xs0[t] = Xb[(s * NL) * NF + t * 16 + lo];
      sYg[t] = 0.f; gPBv[t] = 0.f;
    }
    float gy[9];
#pragma unroll
    for (int l = 0; l < 9; ++l) {
      float gyl = 0.f;
#pragma unroll
      for (int t = 0; t < 4; ++t) {
        int f = t * 16 + lo;
        float gmv = Gb[(d * NL + l) * NF + f];
        float xsl = Xb[(s * NL + l) * NF + f];
        sYg[t] += gmv * Yl[l];
        gPBv[t] += gmv * xsl;
        gyl += gmv * accA[t][v] * xs0[t];
        atomicAdd(&gXb[(s * NL + l) * NF + f], gmv * accB[t][v]);
      }
      gy[l] = gyl;
    }
#pragma unroll
    for (int t = 0; t < 4; ++t) {
      int f = t * 16 + lo;
      atomicAdd(&gXb[(s * NL) * NF + f], sYg[t] * accA[t][v]);
      sPA[wslot][er][f] = sYg[t] * xs0[t];
      sPB[wslot][er][f] = gPBv[t];
    }
#pragma unroll
    for (int l = 0; l < 9; ++l) {
      float r = gy[l];
      r += __shfl_xor(r, 1, 32); r += __shfl_xor(r, 2, 32);
      r += __shfl_xor(r, 4, 32); r += __shfl_xor(r, 8, 32);
      if (lo == 0) gYe[((size_t)b * NE + e) * NL + l] += r;
    }
  }
  __syncthreads();
#pragma unroll
  for (int k8 = 0; k8 < 8; ++k8) {
    int k = hi * 8 + k8;
    float a = 0.f;
    for (int f = 0; f < NF; ++f)
      a += sPA[wslot][lo][f] * WAi[k * NF + f] + sPB[wslot][lo][f] * WBi[k * NF + f];
    gRe[((size_t)b * NE + et * 16 + lo) * NK + k] += a;
  }
}

// geometry backward: g_Y, g_R -> forces
__global__ void __launch_bounds__(256) k_geom_bwd(
    const float* __restrict__ pos, const int* __restrict__ srcI,
    const int* __restrict__ dstI, const float* __restrict__ gYe,
    const float* __restrict__ gRe, float* __restrict__ Fout) {
  int idx = blockIdx.x * blockDim.x + threadIdx.x;
  if (idx >= NB * NE) return;
  int b = idx / NE, e = idx % NE;
  int s = srcI[e], d = dstI[e];
  const float* pb = pos + (size_t)b * NA * 3;
  float dx = pb[s * 3 + 0] - pb[d * 3 + 0];
  float dy = pb[s * 3 + 1] - pb[d * 3 + 1];
  float dz = pb[s * 3 + 2] - pb[d * 3 + 2];
  float r = sqrtf(dx * dx + dy * dy + dz * dz + 1e-12f);
  float inv = 1.f / r;
  float ux = dx * inv, uy = dy * inv, uz = dz * inv;
  const float* gY = gYe + (size_t)idx * NL;
  const float* gR = gRe + (size_t)idx * NK;
  float gux = gY[1] + gY[4] * SQ3F * uy + gY[7] * SQ3F * uz + gY[8] * SQ3F * ux;
  float guy = gY[2] + gY[4] * SQ3F * ux + gY[5] * SQ3F * uz - gY[8] * SQ3F * uy;
  float guz = gY[3] + gY[5] * SQ3F * uy + gY[6] * 3.f * uz + gY[7] * SQ3F * ux;
  float w = 1.f / (1.f + r), om = 1.f - w;
  float t = r * 0.2f; t = t * t;
  float cut = 0.f, dcut = 0.f;
  if (t < 1.f) {
    float den = fmaxf(1.f - t, 1e-12f);
    cut = expf(1.f - 1.f / den);
    if (1.f - t > 1e-12f) dcut = -cut * (2.f * r / 25.f) / (den * den);
  }
  float wp[16], op[16];
  wp[0] = 1.f; op[0] = 1.f;
#pragma unroll
  for (int k = 1; k < 16; ++k) { wp[k] = wp[k - 1] * w; op[k] = op[k - 1] * om; }
  float dwdr = -w * w;
  float g_r = 0.f;
#pragma unroll
  for (int k = 0; k < 16; ++k) {
    float radk = c_BIN[k] * wp[k] * op[15 - k];
    float t1 = (k > 0) ? (float)k * wp[k - 1] * op[15 - k] : 0.f;
    float t2 = (k < 15) ? (float)(15 - k) * wp[k] * op[14 - k] : 0.f;
    float dradw = c_BIN[k] * (t1 - t2);
    g_r += gR[k] * (dradw * dwdr * cut + radk * dcut);
  }
  float dot = gux * ux + guy * uy + guz * uz;
  float gdx = (gux - dot * ux) * inv + g_r * ux;
  float gdy = (guy - dot * uy) * inv + g_r * uy;
  float gdz = (guz - dot * uz) * inv + g_r * uz;
  // F = -dE/dpos : dE/dpos[src] += gdisp, dE/dpos[dst] -= gdisp
  atomicAdd(&Fout[((size_t)b * NA + s) * 3 + 0], -gdx);
  atomicAdd(&Fout[((size_t)b * NA + s) * 3 + 1], -gdy);
  atomicAdd(&Fout[((size_t)b * NA + s) * 3 + 2], -gdz);
  atomicAdd(&Fout[((size_t)b * NA + d) * 3 + 0], gdx);
  atomicAdd(&Fout[((size_t)b * NA + d) * 3 + 1], gdy);
  atomicAdd(&Fout[((size_t)b * NA + d) * 3 + 2], gdz);
}

// ---------------------------------------------------------------------------
extern "C" void kernel_launch(void* const* d_in, const int* in_sizes, int n_in,
                              void* d_out, int out_size, void* d_ws, size_t ws_size,
                              hipStream_t stream) {
  const int* features = (const int*)d_in[0];
  const float* positions = (const float*)d_in[1];
  const float* embed = (const float*)d_in[2];
  const float* WA = (const float*)d_in[3];
  const float* WB = (const float*)d_in[4];
  const float* W1 = (const float*)d_in[5];
  const float* b1 = (const float*)d_in[6];
  const float* W2 = (const float*)d_in[7];
  const float* W1s = (const float*)d_in[8];
  const float* b1s = (const float*)d_in[9];
  const float* W2s = (const float*)d_in[10];
  const float* w_out = (const float*)d_in[11];
  const float* ebias = (const float*)d_in[12];
  const int* dstI = (const int*)d_in[13];
  const int* srcI = (const int*)d_in[14];
  float* out = (float*)d_out;
  float* Eout = out;           // [8]
  float* Fout = out + NB;      // [8,128,3]

  const size_t BNLF = (size_t)NB * NA * NL * NF;   // 589824
  const size_t BNF = (size_t)NB * NA * NF;         // 65536
  const size_t BE9 = (size_t)NB * NE * NL;
  const size_t BE16 = (size_t)NB * NE * NK;

  float* w = (float*)d_ws;
  size_t o = 0;
  auto alloc = [&](size_t n) { float* p = w + o; o += n; return p; };
  float* Ybuf = alloc(BE9);
  float* Rbuf = alloc(BE16);
  float* X0 = alloc(BNLF);
  float* X1 = alloc(BNLF);
  float* X2 = alloc(BNLF);
  float* Y1a = alloc(BNLF);
  float* Y1b = alloc(BNLF);
  float* Gbuf = alloc(BNLF);
  float* GYb = alloc(BNLF);
  float* ysum = alloc(BNLF);
  float* ysum0 = alloc(BNF);
  float* Abuf = alloc(BNF);
  float* H1buf = alloc(BNF);
  float* GAbuf = alloc(BNF);
  float* Gh0buf = alloc(BNF);
  float* gh1 = alloc(64);
  float* gX0 = alloc(BNLF);
  float* gX1 = alloc(BNLF);
  float* gX2 = alloc(BNLF);
  float* gYe = alloc(BE9);
  float* gRe = alloc(BE16);
  (void)ws_size; (void)n_in; (void)in_sizes; (void)out_size;

  const dim3 blk256(256), blk128(128);
  const int edgeBlocks = (NB * NE + 255) / 256;
  const int waveBlocks = (NB * NET) / 4;       // 2032 blocks, 4 waves each
  const int atomBlocks = (NB * NL * 8) / 4;    // 144
  const int mlpBlocks = (NB * 8) / 4;          // 16
  const int ewBNF = (int)((BNF + 255) / 256);
  const int ewBNLF = (int)((BNLF + 255) / 256);

  hipMemsetAsync(d_out, 0, sizeof(float) * (NB + (size_t)NB * NA * 3), stream);

  k_edge_basis<<<edgeBlocks, blk256, 0, stream>>>(positions, srcI, dstI, Ybuf, Rbuf);
  k_embed<<<ewBNLF, blk256, 0, stream>>>(features, embed, X0);

  float* Xs[3] = {X0, X1, X2};
  float* Y1s[2] = {Y1a, Y1b};
  for (int i = 0; i < 2; ++i) {
    hipMemsetAsync(ysum, 0, BNLF * sizeof(float), stream);
    k_msg_fwd<<<waveBlocks, blk128, 0, stream>>>(Rbuf, Ybuf, Xs[i], WA + i * NK * NF,
                                                 WB + i * NK * NF, srcI, dstI, ysum);
    k_atom_fwd1<<<atomBlocks, blk128, 0, stream>>>(Xs[i], ysum, W1 + (size_t)i * 3 * NF * NF,
                                                   b1 + i * NF, Y1s[i]);
    k_gate<<<ewBNF, blk256, 0, stream>>>(Y1s[i], Gbuf);
    k_atom_fwd2<<<atomBlocks, blk128, 0, stream>>>(Gbuf, Xs[i], W2 + (size_t)i * 3 * NF * NF,
                                                   Xs[i + 1]);
  }

  hipMemsetAsync(ysum0, 0, BNF * sizeof(float), stream);
  k_msg_final_fwd<<<waveBlocks, blk128, 0, stream>>>(Rbuf, Ybuf, X2, WA + 2 * NK * NF,
                                                     WB + 2 * NK * NF, srcI, dstI, ysum0);
  k_mlp1<<<mlpBlocks, blk128, 0, stream>>>(X2, ysum0, W1s, b1s, Abuf);
  k_silu<<<ewBNF, blk256, 0, stream>>>(Abuf, H1buf);
  k_mlp2<<<mlpBlocks, blk128, 0, stream>>>(H1buf, W2s, X2, w_out, ebias, features, Eout);

  // ---- backward ----
  k_gh1<<<1, 64, 0, stream>>>(w_out, W2s, gh1);
  k_gax<<<ewBNF, blk256, 0, stream>>>(Abuf, gh1, GAbuf);
  hipMemsetAsync(gX2, 0, BNLF * sizeof(float), stream);
  k_mlp_bwd<<<mlpBlocks, blk128, 0, stream>>>(GAbuf, W1s, w_out, Gh0buf, gX2);
  hipMemsetAsync(gYe, 0, BE9 * sizeof(float), stream);
  hipMemsetAsync(gRe, 0, BE16 * sizeof(float), stream);
  k_msg_final_bwd<<<waveBlocks, blk128, 0, stream>>>(Rbuf, Ybuf, X2, WA + 2 * NK * NF,
                                                     WB + 2 * NK * NF, srcI, dstI, Gh0buf,
                                                     gX2, gYe, gRe);

  float* gXs[3] = {gX0, gX1, gX2};
  for (int i = 1; i >= 0; --i) {
    k_atom_bwd2<<<atomBlocks, blk128, 0, stream>>>(gXs[i + 1], W2 + (size_t)i * 3 * NF * NF, Gbuf);
    k_gate_bwd<<<ewBNF, blk256, 0, stream>>>(Y1s[i], Gbuf);
    k_atom_bwd1<<<atomBlocks, blk128, 0, stream>>>(Gbuf, W1 + (size_t)i * 3 * NF * NF,
                                                   gXs[i + 1], GYb, gXs[i]);
    k_msg_bwd<<<waveBlocks, blk128, 0, stream>>>(Rbuf, Ybuf, Xs[i], WA + i * NK * NF,
                                                 WB + i * NK * NF, srcI, dstI, GYb,
                                                 gXs[i], gYe, gRe);
  }

  k_geom_bwd<<<edgeBlocks, blk256, 0, stream>>>(positions, srcI, dstI, gYe, gRe, Fout);
}